// PixelUnshuffle_53841710023129
// MI455X (gfx1250) — compile-verified
//
#include <hip/hip_runtime.h>

// PixelUnshuffle k=2 on (B=16, C=32, H=512, W=512) fp32.
// out[b, c*4 + y*2 + x, h, w] = in[b, c, 2h+y, 2w+x]
//
// Pure data movement: ~1.07 GB total traffic -> HBM-bound (~46 us at 23.3 TB/s).
// Strategy: per thread, one 8-float row segment via 2x b128 NT loads; split by
// column parity into two float4s that are contiguous in two adjacent output
// channels; 2x b128 NT stores. Fully coalesced on both sides in wave32.

typedef float f4 __attribute__((ext_vector_type(4)));

__global__ __launch_bounds__(256) void pixel_unshuffle_k2_kernel(
    const float* __restrict__ in, float* __restrict__ out) {
  // Compile-time geometry (from the reference).
  constexpr unsigned W    = 512;            // input width
  constexpr unsigned H    = 512;            // input height
  constexpr unsigned HH   = H / 2;          // output height (256)
  constexpr unsigned WW   = W / 2;          // output width  (256)
  constexpr unsigned SEGS = W / 8;          // 8-float segments per input row (64)

  const unsigned tid = blockIdx.x * blockDim.x + threadIdx.x;

  const unsigned seg = tid & (SEGS - 1);    // which 8-float segment in the row
  const unsigned row = tid >> 6;            // flat row index over (B*C*H)

  const unsigned y  = row & 1;              // sub-row parity (k_y)
  const unsigned hh = (row & (H - 1)) >> 1; // output h
  const unsigned bc = row >> 9;             // b*C + c  (H == 512 == 2^9)

  // ---- load 8 consecutive input floats (two b128, non-temporal) ----
  const f4* src = reinterpret_cast<const f4*>(in + (size_t)row * W + (size_t)seg * 8);
  const f4 a = __builtin_nontemporal_load(src);
  const f4 b = __builtin_nontemporal_load(src + 1);

  // Parity split: even columns -> out channel (bc*4 + 2y), odd -> (bc*4 + 2y + 1).
  f4 even, odd;
  even.x = a.x; even.y = a.z; even.z = b.x; even.w = b.z;
  odd.x  = a.y; odd.y  = a.w; odd.z  = b.y; odd.w  = b.w;

  // Output flat index: ((bc*4 + 2y + x) * HH + hh) * WW + 4*seg
  const size_t obase =
      ((size_t)(bc * 4u + y * 2u) * HH + hh) * WW + (size_t)seg * 4u;

  f4* o0 = reinterpret_cast<f4*>(out + obase);                        // x = 0 channel
  f4* o1 = reinterpret_cast<f4*>(out + obase + (size_t)HH * WW);      // x = 1 channel
  __builtin_nontemporal_store(even, o0);
  __builtin_nontemporal_store(odd,  o1);
}

extern "C" void kernel_launch(void* const* d_in, const int* in_sizes, int n_in,
                              void* d_out, int out_size, void* d_ws, size_t ws_size,
                              hipStream_t stream) {
  (void)in_sizes; (void)n_in; (void)out_size; (void)d_ws; (void)ws_size;

  const float* in  = (const float*)d_in[0];
  float*       out = (float*)d_out;

  // B*C*H rows = 16*32*512 = 262,144; 64 segments each -> 16,777,216 threads.
  constexpr unsigned kThreads = 16u * 32u * 512u * (512u / 8u);
  constexpr unsigned kBlock   = 256u;                 // 8 waves on CDNA5 wave32
  constexpr unsigned kGrid    = kThreads / kBlock;    // 65,536 blocks

  pixel_unshuffle_k2_kernel<<<kGrid, kBlock, 0, stream>>>(in, out);
}